// SparseToDenseDirect_22935125361020
// MI455X (gfx1250) — compile-verified
//
#include <hip/hip_runtime.h>
#include <stdint.h>

// Problem constants (match reference).
#define B_    4
#define C_    128
#define Y_    512
#define X_    512
#define YX_   (Y_ * X_)            // 262144 = 2^18
#define CELLS_ (B_ * YX_)          // 1048576 = 2^20
#define OUT_ELEMS_ ((long long)CELLS_ * C_)   // 134217728

#define TILE_CELLS 64
#define LDS_PITCH  129             // 128 dwords + 1 pad dword (TDM pad) -> conflict-free transpose

#if defined(__has_builtin)
#if __has_builtin(__builtin_amdgcn_tensor_load_to_lds)
#define HAVE_TDM 1
#endif
#if __has_builtin(__builtin_amdgcn_s_wait_tensorcnt)
#define HAVE_WAIT_TENSOR 1
#endif
#endif

typedef unsigned int u32x4 __attribute__((ext_vector_type(4)));
typedef int          i32x8 __attribute__((ext_vector_type(8)));
typedef int          i32x4 __attribute__((ext_vector_type(4)));

// Monotonic order-preserving encode: larger float <-> larger unsigned key.
// key 0 is reserved: no real float encodes to 0 (only ~0xFFFFFFFF = -NaN would).
__device__ __forceinline__ unsigned encodeKey(float f) {
    unsigned b = __float_as_uint(f);
    return (b & 0x80000000u) ? ~b : (b | 0x80000000u);
}

__device__ __forceinline__ float decodeKey(unsigned k) {
    if (k == 0u) return 0.0f;  // untouched cell -> exactly 0 (reference semantics)
    unsigned bits = (k & 0x80000000u) ? (k & 0x7FFFFFFFu) : ~k;
    return __uint_as_float(bits);
}

// ---------------------------------------------------------------------------
// Kernel 1: zero an encoded buffer (uint4 vectorized, grid-stride).
// ---------------------------------------------------------------------------
__global__ void zero_u32_kernel(uint4* __restrict__ p, long long n4) {
    long long i      = (long long)blockIdx.x * blockDim.x + threadIdx.x;
    long long stride = (long long)gridDim.x * blockDim.x;
    uint4 z; z.x = 0u; z.y = 0u; z.z = 0u; z.w = 0u;
    for (; i < n4; i += stride) p[i] = z;
}

// ---------------------------------------------------------------------------
// Kernel 2a: scatter-max into cell-major staging [cells, C] (coalesced atomics:
// one voxel's 128 channels -> 512 contiguous bytes).
// ---------------------------------------------------------------------------
__global__ __launch_bounds__(256) void scatter_bev_ws(
        const float* __restrict__ feats, const int* __restrict__ coords,
        unsigned* __restrict__ enc, int nVox) {
    int vox = blockIdx.x * 2 + (threadIdx.x >> 7);
    int c   = threadIdx.x & 127;
    if (vox >= nVox) return;
    int b = coords[vox * 4 + 0];
    int y = coords[vox * 4 + 2];
    int x = coords[vox * 4 + 3];
    unsigned cell = ((unsigned)b * Y_ + (unsigned)y) * X_ + (unsigned)x;
    float f = feats[(size_t)vox * C_ + c];
    atomicMax(&enc[(size_t)cell * C_ + c], encodeKey(f));
}

// ---------------------------------------------------------------------------
// Kernel 2b (fallback if ws too small): scatter directly into [B,C,Y,X].
// ---------------------------------------------------------------------------
__global__ __launch_bounds__(256) void scatter_bev_direct(
        const float* __restrict__ feats, const int* __restrict__ coords,
        unsigned* __restrict__ enc, int nVox) {
    int vox = blockIdx.x * 2 + (threadIdx.x >> 7);
    int c   = threadIdx.x & 127;
    if (vox >= nVox) return;
    int b = coords[vox * 4 + 0];
    int y = coords[vox * 4 + 2];
    int x = coords[vox * 4 + 3];
    size_t idx = (((size_t)b * C_ + c) * Y_ + y) * X_ + x;
    float f = feats[(size_t)vox * C_ + c];
    atomicMax(&enc[idx], encodeKey(f));
}

// ---------------------------------------------------------------------------
// Kernel 3a: tiled transpose [cells, C] -> [B, C, Y, X] with decode.
// TDM loads a 64x128-dword tile into LDS with 1-dword padding per 128 dwords
// (pitch 129 -> bank-conflict-free transposed reads, since 129 mod 64 == 1).
// ---------------------------------------------------------------------------
__global__ __launch_bounds__(256) void transpose_decode(
        const unsigned* __restrict__ enc, float* __restrict__ out) {
    __shared__ unsigned tile[TILE_CELLS * LDS_PITCH];
    const unsigned t         = threadIdx.x;
    const unsigned cellStart = blockIdx.x * TILE_CELLS;

#if defined(HAVE_TDM)
    if ((t >> 5) == 0) {   // one wave issues the DMA (TDM ignores EXEC; once per wave)
        unsigned long long gaddr =
            (unsigned long long)(const void*)(enc + (size_t)cellStart * C_);
        // Low 32 bits of a generic shared pointer == LDS byte address (flat-LDS aperture).
        unsigned ldsOff = (unsigned)(unsigned long long)(const void*)&tile[0];

        // D# group 0: count=1 | lds_addr | global_addr[56:0] | type=2
        u32x4 g0 = { 1u,
                     ldsOff,
                     (unsigned)gaddr,
                     (((unsigned)(gaddr >> 32)) & 0x01FFFFFFu) | 0x80000000u };

        // D# group 1: data_size=4B(2), pad_enable=1, pad_interval=6 (128 dwords),
        // pad_amount=0 (1 dword). 2-D tensor: dim0=C_, dim1=CELLS_, stride0=C_,
        // tile 128 x 64.
        i32x8 g1 = {
            (int)((2u << 16) | (1u << 20) | (6u << 22)),            // w0
            (int)(((unsigned)C_ & 0xFFFFu) << 16),                   // w1: tensor_dim0 lo16
            (int)(((unsigned)C_ >> 16) |
                  (((unsigned)CELLS_ & 0xFFFFu) << 16)),             // w2: dim0 hi16 | dim1 lo16
            (int)(((unsigned)CELLS_ >> 16) | ((unsigned)C_ << 16)),  // w3: dim1 hi16 | tile_dim0
            (int)TILE_CELLS,                                         // w4: tile_dim1 | tile_dim2=0
            (int)C_,                                                 // w5: tensor_dim0_stride lo32
            0,                                                       // w6
            0 };                                                     // w7
        i32x4 gz4 = { 0, 0, 0, 0 };                                  // groups 2/3: 2-D tensor
        i32x8 gz8 = { 0, 0, 0, 0, 0, 0, 0, 0 };                      // extra group (unused)
        __builtin_amdgcn_tensor_load_to_lds(g0, g1, gz4, gz4, gz8, 0);
#if defined(HAVE_WAIT_TENSOR)
        __builtin_amdgcn_s_wait_tensorcnt(0);
#else
        asm volatile("s_wait_tensorcnt 0" ::: "memory");
#endif
    }
    __syncthreads();
#else
    // Fallback: cooperative coalesced load into padded LDS tile.
    for (unsigned e = t; e < TILE_CELLS * C_; e += 256u) {
        unsigned i = e >> 7, c = e & 127u;
        tile[i * LDS_PITCH + c] = enc[(size_t)(cellStart + i) * C_ + c];
    }
    __syncthreads();
#endif

    // Transposed, decoded, float4-coalesced stores.
    const unsigned bb  = cellStart >> 18;           // batch (tiles never straddle batches)
    const unsigned yx0 = cellStart & (YX_ - 1);
    const unsigned i0  = (t & 15u) * 4u;            // cell offset within tile
    for (unsigned k = 0; k < 8; ++k) {
        unsigned c = (t >> 4) + k * 16u;            // 0..127
        float4 v;
        v.x = decodeKey(tile[(i0 + 0u) * LDS_PITCH + c]);
        v.y = decodeKey(tile[(i0 + 1u) * LDS_PITCH + c]);
        v.z = decodeKey(tile[(i0 + 2u) * LDS_PITCH + c]);
        v.w = decodeKey(tile[(i0 + 3u) * LDS_PITCH + c]);
        size_t outIdx = (((size_t)(bb * C_ + c)) << 18) + yx0 + i0;
        *(float4*)(out + outIdx) = v;
    }
}

// ---------------------------------------------------------------------------
// Kernel 3b (fallback path): in-place decode of d_out.
// ---------------------------------------------------------------------------
__global__ void decode_inplace(unsigned* __restrict__ buf, long long n4) {
    long long i      = (long long)blockIdx.x * blockDim.x + threadIdx.x;
    long long stride = (long long)gridDim.x * blockDim.x;
    uint4*  p4 = (uint4*)buf;
    float4* f4 = (float4*)buf;
    for (; i < n4; i += stride) {
        uint4 u = p4[i];
        float4 f;
        f.x = decodeKey(u.x); f.y = decodeKey(u.y);
        f.z = decodeKey(u.z); f.w = decodeKey(u.w);
        f4[i] = f;
    }
}

extern "C" void kernel_launch(void* const* d_in, const int* in_sizes, int n_in,
                              void* d_out, int out_size, void* d_ws, size_t ws_size,
                              hipStream_t stream) {
    const float* feats  = (const float*)d_in[0];
    const int*   coords = (const int*)d_in[1];
    const int    nVox   = in_sizes[1] / 4;

    const long long n4 = OUT_ELEMS_ / 4;              // uint4 count for 512 MB
    const int zeroBlocks    = 4096;
    const int scatterBlocks = (nVox + 1) / 2;         // 2 voxels per 256-thread block
    const size_t needWs = (size_t)OUT_ELEMS_ * 4ull;  // 512 MB staging

    if (ws_size >= needWs) {
        unsigned* enc = (unsigned*)d_ws;
        zero_u32_kernel<<<zeroBlocks, 256, 0, stream>>>((uint4*)enc, n4);
        scatter_bev_ws<<<scatterBlocks, 256, 0, stream>>>(feats, coords, enc, nVox);
        transpose_decode<<<CELLS_ / TILE_CELLS, 256, 0, stream>>>(enc, (float*)d_out);
    } else {
        unsigned* enc = (unsigned*)d_out;
        zero_u32_kernel<<<zeroBlocks, 256, 0, stream>>>((uint4*)enc, n4);
        scatter_bev_direct<<<scatterBlocks, 256, 0, stream>>>(feats, coords, enc, nVox);
        decode_inplace<<<zeroBlocks, 256, 0, stream>>>(enc, n4);
    }
    (void)n_in; (void)out_size;
}